// GraphEncoder_71966472012662
// MI455X (gfx1250) — compile-verified
//
#include <hip/hip_runtime.h>
#include <hip/hip_bf16.h>

#define N_NODES 128
#define D_DIM   256
#define T_HEAD  512
#define LDPAD   132   // LDS row stride in floats (pad to kill bank conflicts on column reads)
#define NWAVES  8

typedef __attribute__((ext_vector_type(2))) float v2f;
typedef __attribute__((ext_vector_type(8))) float v8f;

__device__ __forceinline__ float wredf(float v) {
#pragma unroll
  for (int m = 16; m > 0; m >>= 1) v += __shfl_xor(v, m, 32);
  return v;
}
__device__ __forceinline__ int wredi(int v) {
#pragma unroll
  for (int m = 16; m > 0; m >>= 1) v += __shfl_xor(v, m, 32);
  return v;
}

__global__ __launch_bounds__(256) void coarsen_graph_kernel(
    const float* __restrict__ x, const float* __restrict__ adj,
    const int* __restrict__ head, const float* __restrict__ w,
    const float* __restrict__ bias, float* __restrict__ outE,
    float* __restrict__ outA)
{
  extern __shared__ float smem[];
  float* sAdj = smem;                          // 128 x LDPAD
  float* sS   = sAdj + N_NODES * LDPAD;        // 128 x LDPAD
  float* sT   = sS   + N_NODES * LDPAD;        // 128 x LDPAD  (T1 = S^T adj)
  float* sLin = sT   + N_NODES * LDPAD;        // 128 (x @ w)
  float* sCs  = sLin + N_NODES;                // 128 col scale d_j
  float* sRs  = sCs  + N_NODES;                // 128 row scale mask_i * d_i
  float* sAl  = sRs  + N_NODES;                // 128 alpha
  float* sCf  = sAl  + N_NODES;                // 128 (t_j, later d_j * cut_alpha_j)
  float* sMisc = sCf + N_NODES;                // [0] = cut value
  int*   sInt  = (int*)(sMisc + 4);            // [0] = n_uniq
  int*   sFlag = sInt + 4;                     // 128 presence flags

  const int b    = blockIdx.x;
  const int tid  = threadIdx.x;
  const int lane = tid & 31;
  const int wave = tid >> 5;
  const float* xb = x   + (size_t)b * N_NODES * D_DIM;
  const float* ab = adj + (size_t)b * N_NODES * N_NODES;
  const float bias0 = bias[0];

  // ---- Phase 1: stage adj into LDS, degrees, x@w, zero flags ----
  for (int r = wave; r < N_NODES; r += NWAVES) {
    float4 v = ((const float4*)(ab + (size_t)r * N_NODES))[lane];
    ((float4*)(sAdj + r * LDPAD))[lane] = v;
    float s = wredf(v.x + v.y + v.z + v.w);           // adj row sum
    if (lane == 0) {
      float deg = s + 1.0f;                           // A = adj + I
      float dis = rsqrtf(fmaxf(deg, 1.0f));           // clip(deg,1)^-0.5
      sCs[r] = dis;
      sRs[r] = (s > 0.0f) ? dis : 0.0f;               // row_mask folded in
    }
  }
  for (int r = wave; r < N_NODES; r += NWAVES) {
    const float* xr = xb + (size_t)r * D_DIM;
    float s = 0.0f;
    for (int j = lane; j < D_DIM; j += 32) s += xr[j] * w[j];
    s = wredf(s);
    if (lane == 0) sLin[r] = s;
  }
  if (tid < N_NODES) sFlag[tid] = 0;
  __syncthreads();

  // ---- Phase 2: t_j = d_j * lin_j ; head presence flags ----
  if (tid < N_NODES) sCf[tid] = sCs[tid] * sLin[tid];
  const int* hb = head + (size_t)b * T_HEAD;
  for (int t = tid; t < T_HEAD; t += 256) sFlag[hb[t] & (N_NODES - 1)] = 1;
  __syncthreads();

  // ---- Phase 3: out2 = norm_adj @ lin + bias ; alpha = sigmoid(out2^2) ----
  for (int r = wave; r < N_NODES; r += NWAVES) {
    const float* arow = sAdj + r * LDPAD;
    float s = 0.0f;
    for (int j = lane; j < N_NODES; j += 32) s += arow[j] * sCf[j];
    s = wredf(s);
    if (lane == 0) {
      s += sCf[r];                                    // identity of A = adj + I
      float o = sRs[r] * s + bias0;
      float z = o * o;
      sAl[r] = 1.0f / (1.0f + expf(-z));
    }
  }
  if (wave == 0) {
    int c = sFlag[lane] + sFlag[lane + 32] + sFlag[lane + 64] + sFlag[lane + 96];
    c = wredi(c);
    if (lane == 0) sInt[0] = c;
  }
  __syncthreads();

  // ---- Phase 4: rank-select the k-th largest alpha (exact, tie-correct) ----
  const int nuniq = sInt[0];
  int kk  = (int)ceilf((float)nuniq * 0.1f) + 1;
  int idx = kk - 1; idx = idx < 0 ? 0 : (idx > N_NODES - 1 ? N_NODES - 1 : idx);
  if (tid < N_NODES) {
    float at = sAl[tid];
    int g = 0, e = 0;
#pragma unroll 4
    for (int j = 0; j < N_NODES; ++j) {
      float aj = sAl[j];
      g += (aj > at);
      e += (aj == at);
    }
    if (g <= idx && idx < g + e) sMisc[0] = at;       // descending-sorted[idx]
  }
  __syncthreads();
  const float cut = (nuniq > 1) ? sMisc[0] : 0.0f;
  if (tid < N_NODES) {
    float ca = fmaxf(sAl[tid] + 1e-7f - cut, 0.0f);   // relu(alpha + eps - cut)
    sCf[tid] = sCs[tid] * ca;                         // column factor d_j * ca_j
  }
  __syncthreads();

  // ---- Phase 5: S rows = rowL1norm( rscale_i * (adj+I)_ij * colf_j ) ----
  for (int r = wave; r < N_NODES; r += NWAVES) {
    float rs = sRs[r];
    float v[4]; float asum = 0.0f;
#pragma unroll
    for (int q = 0; q < 4; ++q) {
      int j = lane + 32 * q;
      float a = sAdj[r * LDPAD + j] + ((j == r) ? 1.0f : 0.0f);
      float t = rs * a * sCf[j];
      v[q] = t; asum += fabsf(t);
    }
    asum = wredf(asum);
    float inv = 1.0f / fmaxf(asum, 1e-12f);
#pragma unroll
    for (int q = 0; q < 4; ++q) sS[r * LDPAD + lane + 32 * q] = v[q] * inv;
  }
  __syncthreads();

  // ---- Phase 6: WMMA GEMMs (f32 16x16x4) ----
  const int cL = lane & 15;     // column within fragment
  const int kh = lane >> 4;     // K-half select (lanes 16..31 hold K=2,3 / M+8)

  // E = S^T @ x   [128 x 256]
  float* eb = outE + (size_t)b * N_NODES * D_DIM;
  for (int t = wave; t < 8 * 16; t += NWAVES) {
    const int m = t >> 4, n = t & 15;
    v8f acc = {0.f,0.f,0.f,0.f,0.f,0.f,0.f,0.f};
    const int colA = m * 16 + cL;
    const float* bcol = xb + n * 16 + cL;
#pragma unroll 4
    for (int k0 = 0; k0 < N_NODES; k0 += 4) {
      const int kr = k0 + 2 * kh;
      v2f A, Bf;
      A.x  = sS[kr * LDPAD + colA];                   // (S^T)[m..,k] = S[k][m..]
      A.y  = sS[(kr + 1) * LDPAD + colA];
      Bf.x = bcol[(size_t)kr * D_DIM];                // x[k][n..] (L2-hot)
      Bf.y = bcol[(size_t)(kr + 1) * D_DIM];
      acc = __builtin_amdgcn_wmma_f32_16x16x4_f32(false, A, false, Bf,
                                                  (short)0, acc, false, false);
    }
    const int col = n * 16 + cL;
#pragma unroll
    for (int i = 0; i < 8; ++i)
      eb[(size_t)(m * 16 + i + 8 * kh) * D_DIM + col] = acc[i];
  }

  // T1 = S^T @ adj   [128 x 128] -> LDS
  for (int t = wave; t < 8 * 8; t += NWAVES) {
    const int m = t >> 3, n = t & 7;
    v8f acc = {0.f,0.f,0.f,0.f,0.f,0.f,0.f,0.f};
    const int colA = m * 16 + cL;
    const int colB = n * 16 + cL;
#pragma unroll 4
    for (int k0 = 0; k0 < N_NODES; k0 += 4) {
      const int kr = k0 + 2 * kh;
      v2f A, Bf;
      A.x  = sS[kr * LDPAD + colA];
      A.y  = sS[(kr + 1) * LDPAD + colA];
      Bf.x = sAdj[kr * LDPAD + colB];
      Bf.y = sAdj[(kr + 1) * LDPAD + colB];
      acc = __builtin_amdgcn_wmma_f32_16x16x4_f32(false, A, false, Bf,
                                                  (short)0, acc, false, false);
    }
#pragma unroll
    for (int i = 0; i < 8; ++i)
      sT[(m * 16 + i + 8 * kh) * LDPAD + colB] = acc[i];
  }
  __syncthreads();

  // new_adj = T1 @ S   [128 x 128]
  float* nb = outA + (size_t)b * N_NODES * N_NODES;
  for (int t = wave; t < 8 * 8; t += NWAVES) {
    const int m = t >> 3, n = t & 7;
    v8f acc = {0.f,0.f,0.f,0.f,0.f,0.f,0.f,0.f};
    const float* arow = sT + (size_t)(m * 16 + cL) * LDPAD;   // T1 row-major
    const int colB = n * 16 + cL;
#pragma unroll 4
    for (int k0 = 0; k0 < N_NODES; k0 += 4) {
      const int kr = k0 + 2 * kh;
      v2f A, Bf;
      A.x  = arow[kr];
      A.y  = arow[kr + 1];
      Bf.x = sS[kr * LDPAD + colB];
      Bf.y = sS[(kr + 1) * LDPAD + colB];
      acc = __builtin_amdgcn_wmma_f32_16x16x4_f32(false, A, false, Bf,
                                                  (short)0, acc, false, false);
    }
#pragma unroll
    for (int i = 0; i < 8; ++i)
      nb[(size_t)(m * 16 + i + 8 * kh) * N_NODES + colB] = acc[i];
  }
}

extern "C" void kernel_launch(void* const* d_in, const int* in_sizes, int n_in,
                              void* d_out, int out_size, void* d_ws, size_t ws_size,
                              hipStream_t stream) {
  const float* x    = (const float*)d_in[0];   // [B,N,D]
  const float* adj  = (const float*)d_in[1];   // [B,N,N]
  const int*   head = (const int*)  d_in[2];   // [B,T]
  const float* w    = (const float*)d_in[3];   // [1,D]
  const float* bias = (const float*)d_in[4];   // [1]
  const int B = in_sizes[0] / (N_NODES * D_DIM);
  float* outE = (float*)d_out;                               // [B,N,D]
  float* outA = outE + (size_t)B * N_NODES * D_DIM;          // [B,N,N]
  const size_t smem = (size_t)(3 * N_NODES * LDPAD + 5 * N_NODES + 4) * sizeof(float)
                    + (size_t)(4 + N_NODES) * sizeof(int);   // ~206 KB of the 320 KB WGP LDS
  hipLaunchKernelGGL(coarsen_graph_kernel, dim3(B), dim3(256), smem, stream,
                     x, adj, head, w, bias, outE, outA);
}